// Attention_12292196401159
// MI455X (gfx1250) — compile-verified
//
#include <hip/hip_runtime.h>
#include <hip/hip_bf16.h>

// ---------------------------------------------------------------------------
// Attention step (pointer-generator style), B=32, TK=2048, N=1024, f32.
// Memory-bound: 2x 256MB streamed tensors dominate (~22us at 23.3 TB/s).
// Small GEMMs done in exact f32 via V_WMMA_F32_16X16X4_F32.
// ---------------------------------------------------------------------------

#define B_SZ  32
#define TK_SZ 2048
#define N_SZ  1024

typedef __attribute__((ext_vector_type(2))) float v2f;
typedef __attribute__((ext_vector_type(8))) float v8f;

__device__ __forceinline__ float fast_tanhf(float x) {
#if __has_builtin(__builtin_amdgcn_tanhf)
    return __builtin_amdgcn_tanhf(x);   // v_tanh_f32 on gfx1250
#else
    float e = __expf(2.0f * x);
    return 1.0f - 2.0f / (e + 1.0f);
#endif
}

// ---------------------------------------------------------------------------
// Kernel 1: proj[b][n] = (s_t_hat @ W_dec^T)[b][n] + (c_t_v @ W_ctv^T)[b][n]
//                        + b_dec[n] + b_ctv[n]
// M=32 (2 tiles), N=1024 (64 tiles) -> 128 waves, one 16x16 tile per wave.
// f32 WMMA 16x16x4; both GEMMs chained into the same accumulator.
//
// f32 A 16x4 layout: lanes 0-15 M=0..15, VGPR0/1 = K={0,1} ; lanes 16-31
// hold K={2,3}. B 4x16 mirrors the K split with N across lanes 0-15.
// Since B[k][n] = W[n][k], each lane reads a contiguous float2 of row n of W.
// ---------------------------------------------------------------------------
__global__ __launch_bounds__(256) void proj_gemm_wmma(
    const float* __restrict__ s_t_hat,   // [32,1024]
    const float* __restrict__ c_t_v,     // [32,1024]
    const float* __restrict__ W_dec,     // [1024,1024]
    const float* __restrict__ W_ctv,     // [1024,1024]
    const float* __restrict__ b_dec,     // [1024]
    const float* __restrict__ b_ctv,     // [1024]
    float* __restrict__ proj)            // [32,1024]
{
    const int wave = threadIdx.x >> 5;
    const int lane = threadIdx.x & 31;
    const int tile = blockIdx.x * 8 + wave;   // 0..127
    const int mt = tile >> 6;                 // 0..1
    const int nt = tile & 63;                 // 0..63
    const int m0 = mt * 16, n0 = nt * 16;
    const int hi  = lane >> 4;                // 0: K pair {0,1}, 1: K pair {2,3}
    const int l15 = lane & 15;
    const int m = m0 + l15;
    const int n = n0 + l15;

    const float* arow_dec = s_t_hat + m * N_SZ;
    const float* arow_ctv = c_t_v   + m * N_SZ;
    const float* brow_dec = W_dec   + n * N_SZ;  // B[k][n] = W[n][k]
    const float* brow_ctv = W_ctv   + n * N_SZ;

    v8f acc = {0.f, 0.f, 0.f, 0.f, 0.f, 0.f, 0.f, 0.f};
    for (int k0 = 0; k0 < N_SZ; k0 += 4) {
        const int ko = k0 + 2 * hi;
        v2f a0 = *(const v2f*)(arow_dec + ko);
        v2f b0 = *(const v2f*)(brow_dec + ko);
        acc = __builtin_amdgcn_wmma_f32_16x16x4_f32(
            false, a0, false, b0, (short)0, acc, false, false);
        v2f a1 = *(const v2f*)(arow_ctv + ko);
        v2f b1 = *(const v2f*)(brow_ctv + ko);
        acc = __builtin_amdgcn_wmma_f32_16x16x4_f32(
            false, a1, false, b1, (short)0, acc, false, false);
    }

    const float bias = b_dec[n] + b_ctv[n];
    // C/D layout: VGPR r -> M = r (lanes 0-15) / M = r+8 (lanes 16-31), N = lane&15
#pragma unroll
    for (int r = 0; r < 8; ++r) {
        const int mm = m0 + r + 8 * hi;
        proj[mm * N_SZ + n] = acc[r] + bias;
    }
}

// ---------------------------------------------------------------------------
// Kernel 2: scores[b][t] = sum_n v_w[n] * tanh(enc_feat[b,t,n] + proj[b,n]
//                                              + coverage[b,t] * W_c[n])
// One wave per t-row (4KB streamed per wave, float4 loads). proj/W_c/v_w
// staged in LDS (12KB). Grid (32, 256) = 8192 blocks for full occupancy.
// ---------------------------------------------------------------------------
__global__ __launch_bounds__(256) void scores_kernel(
    const float* __restrict__ encoder_feature,  // [32,2048,1024]
    const float* __restrict__ proj,             // [32,1024]
    const float* __restrict__ coverage,         // [32,2048]
    const float* __restrict__ W_c,              // [1024]
    const float* __restrict__ v_w,              // [1024]
    float* __restrict__ scores)                 // [32,2048]
{
    __shared__ __align__(16) float proj_s[N_SZ];
    __shared__ __align__(16) float wc_s[N_SZ];
    __shared__ __align__(16) float vw_s[N_SZ];

    const int b = blockIdx.x;
    for (int i = threadIdx.x; i < N_SZ; i += 256) {
        proj_s[i] = proj[b * N_SZ + i];
        wc_s[i]   = W_c[i];
        vw_s[i]   = v_w[i];
    }
    __syncthreads();

    const int wave = threadIdx.x >> 5;
    const int lane = threadIdx.x & 31;
    const int t = blockIdx.y * 8 + wave;
    const float cov = coverage[b * TK_SZ + t];

    const float4* ef = (const float4*)(encoder_feature +
                                       ((size_t)b * TK_SZ + t) * N_SZ);
    const float4* ps = (const float4*)proj_s;
    const float4* ws = (const float4*)wc_s;
    const float4* vs = (const float4*)vw_s;

    float sum = 0.f;
#pragma unroll
    for (int j = 0; j < 8; ++j) {
        const int idx = lane + j * 32;        // float4 index 0..255
        float4 x = ef[idx];
        float4 p = ps[idx];
        float4 w = ws[idx];
        float4 v = vs[idx];
        sum += v.x * fast_tanhf(x.x + p.x + cov * w.x);
        sum += v.y * fast_tanhf(x.y + p.y + cov * w.y);
        sum += v.z * fast_tanhf(x.z + p.z + cov * w.z);
        sum += v.w * fast_tanhf(x.w + p.w + cov * w.w);
    }
#pragma unroll
    for (int o = 16; o > 0; o >>= 1) sum += __shfl_xor(sum, o, 32);
    if (lane == 0) scores[b * TK_SZ + t] = sum;
}

// ---------------------------------------------------------------------------
// Kernel 3: masked softmax over t (renormalized), write attn_dist and
// coverage_out directly into d_out. One block of 256 per batch row; each
// thread owns 8 t-values.  softmax(s)*mask renormalized == exp(s-max)*mask
// normalized by its own sum (the first denominator cancels).
// ---------------------------------------------------------------------------
__global__ __launch_bounds__(256) void softmax_kernel(
    const float* __restrict__ scores,    // [32,2048]
    const float* __restrict__ mask,      // [32,2048]
    const float* __restrict__ coverage,  // [32,2048]
    float* __restrict__ attn_out,        // [32,2048]
    float* __restrict__ cov_out)         // [32,2048]
{
    __shared__ float red[8];
    const int b = blockIdx.x;
    const int tid = threadIdx.x;

    float s[8];
    float m = -3.402823466e38f;
#pragma unroll
    for (int j = 0; j < 8; ++j) {
        s[j] = scores[b * TK_SZ + tid + j * 256];
        m = fmaxf(m, s[j]);
    }
#pragma unroll
    for (int o = 16; o > 0; o >>= 1) m = fmaxf(m, __shfl_xor(m, o, 32));
    if ((tid & 31) == 0) red[tid >> 5] = m;
    __syncthreads();
    m = red[0];
#pragma unroll
    for (int i = 1; i < 8; ++i) m = fmaxf(m, red[i]);
    __syncthreads();

    float e[8];
    float lsum = 0.f;
#pragma unroll
    for (int j = 0; j < 8; ++j) {
        const float msk = mask[b * TK_SZ + tid + j * 256];
        e[j] = __expf(s[j] - m) * msk;
        lsum += e[j];
    }
#pragma unroll
    for (int o = 16; o > 0; o >>= 1) lsum += __shfl_xor(lsum, o, 32);
    if ((tid & 31) == 0) red[tid >> 5] = lsum;
    __syncthreads();
    float denom = 0.f;
#pragma unroll
    for (int i = 0; i < 8; ++i) denom += red[i];
    const float inv = 1.0f / denom;

#pragma unroll
    for (int j = 0; j < 8; ++j) {
        const int t = tid + j * 256;
        const float a = e[j] * inv;
        attn_out[b * TK_SZ + t] = a;
        cov_out[b * TK_SZ + t] = coverage[b * TK_SZ + t] + a;
    }
}

// ---------------------------------------------------------------------------
// Kernel 4: partial context sums. Grid (32, 16): block (b, t-chunk of 128)
// streams 512KB of encoder_outputs with float4 loads (each thread owns 4
// consecutive n), attn chunk staged in LDS. Deterministic (no FP atomics).
// ---------------------------------------------------------------------------
__global__ __launch_bounds__(256) void context_partial_kernel(
    const float* __restrict__ encoder_outputs,  // [32,2048,1024]
    const float* __restrict__ attn,             // [32,2048]
    float* __restrict__ partial)                // [32,16,1024]
{
    __shared__ float a_s[128];
    const int b  = blockIdx.x;
    const int tc = blockIdx.y;
    const int t0 = tc * 128;
    if (threadIdx.x < 128)
        a_s[threadIdx.x] = attn[b * TK_SZ + t0 + threadIdx.x];
    __syncthreads();

    const float4* eo = (const float4*)(encoder_outputs +
                                       ((size_t)b * TK_SZ + t0) * N_SZ);
    const int nv = threadIdx.x;   // float4 column 0..255
    float4 acc = make_float4(0.f, 0.f, 0.f, 0.f);
    for (int i = 0; i < 128; ++i) {
        float4 x = eo[(size_t)i * 256 + nv];
        const float a = a_s[i];
        acc.x += a * x.x;
        acc.y += a * x.y;
        acc.z += a * x.z;
        acc.w += a * x.w;
    }
    ((float4*)partial)[(b * 16 + tc) * 256 + nv] = acc;
}

// ---------------------------------------------------------------------------
// Kernel 5: reduce 16 t-chunk partials -> c_t[b][n].
// ---------------------------------------------------------------------------
__global__ __launch_bounds__(256) void context_reduce_kernel(
    const float* __restrict__ partial,  // [32,16,1024]
    float* __restrict__ c_t)            // [32,1024]
{
    const int idx = blockIdx.x * 256 + threadIdx.x;  // 0..32767
    const int b = idx >> 10;
    const int n = idx & (N_SZ - 1);
    float s = 0.f;
#pragma unroll
    for (int i = 0; i < 16; ++i) s += partial[(b * 16 + i) * N_SZ + n];
    c_t[idx] = s;
}

// ---------------------------------------------------------------------------
// Host launcher
// ---------------------------------------------------------------------------
extern "C" void kernel_launch(void* const* d_in, const int* in_sizes, int n_in,
                              void* d_out, int out_size, void* d_ws, size_t ws_size,
                              hipStream_t stream) {
    const float* s_t_hat          = (const float*)d_in[0];
    const float* encoder_outputs  = (const float*)d_in[1];
    const float* encoder_feature  = (const float*)d_in[2];
    const float* enc_padding_mask = (const float*)d_in[3];
    const float* coverage         = (const float*)d_in[4];
    const float* c_t_v            = (const float*)d_in[5];
    const float* W_dec            = (const float*)d_in[6];
    const float* b_dec            = (const float*)d_in[7];
    const float* W_ctv            = (const float*)d_in[8];
    const float* b_ctv            = (const float*)d_in[9];
    const float* W_c              = (const float*)d_in[10];
    const float* v_w              = (const float*)d_in[11];

    // Output layout (flat, return order): c_t [32,1024] | attn [32,2048] | coverage_out [32,2048]
    float* out      = (float*)d_out;
    float* c_t      = out;                      // 32768
    float* attn     = out + B_SZ * N_SZ;        // 65536
    float* cov_out  = attn + B_SZ * TK_SZ;      // 65536

    // Workspace: proj (128KB) | scores (256KB) | partials (2MB)
    float* ws      = (float*)d_ws;
    float* proj    = ws;                        // 32*1024
    float* scores  = proj + B_SZ * N_SZ;        // 32*2048
    float* partial = scores + B_SZ * TK_SZ;     // 32*16*1024

    proj_gemm_wmma<<<16, 256, 0, stream>>>(s_t_hat, c_t_v, W_dec, W_ctv,
                                           b_dec, b_ctv, proj);
    scores_kernel<<<dim3(B_SZ, TK_SZ / 8), 256, 0, stream>>>(
        encoder_feature, proj, coverage, W_c, v_w, scores);
    softmax_kernel<<<B_SZ, 256, 0, stream>>>(scores, enc_padding_mask,
                                             coverage, attn, cov_out);
    context_partial_kernel<<<dim3(B_SZ, 16), 256, 0, stream>>>(
        encoder_outputs, attn, partial);
    context_reduce_kernel<<<(B_SZ * N_SZ) / 256, 256, 0, stream>>>(partial, c_t);
}